// AI4CFD_81020263071752
// MI455X (gfx1250) — compile-verified
//
#include <hip/hip_runtime.h>
#include <cstdint>
#include <cstddef>

// ---------------------------------------------------------------------------
// Problem constants (match the JAX reference)
// ---------------------------------------------------------------------------
constexpr int   NYc = 160;
constexpr int   NXc = 6600;
constexpr int   Nc  = NYc * NXc;          // 1,056,000
constexpr int   PWc = NXc + 2;            // 6602
constexpr int   PHc = NYc + 2;            // 162
constexpr int   NPc = PWc * PHc;          // 1,069,524

constexpr float DTc   = 0.005f;
constexpr float NUc   = 0.01f;            // 1/RE
constexpr float KTc   = 1.0f / 70.0f;     // 1/(RE*PR)
constexpr float INV_DIAG = 0.375f;        // 1/(8/3)
constexpr float INV_HALF_IN = 1.0f / 2501.0f;  // 1/(1+0.5*DT*1e6)
constexpr float INV_FULL_IN = 1.0f / 5001.0f;  // 1/(1+DT*1e6)
constexpr int   COR_Xc = 1800;            // 45*40
constexpr int   COR_Yc = 80;
constexpr int   RAD2c  = 400;             // (40/2)^2

#define KERN __global__ __launch_bounds__(256)

__device__ __forceinline__ bool inside_body(int x, int y) {
    int dx = x - COR_Xc, dy = y - COR_Yc;
    return (x >= COR_Xc) && (dx * dx + dy * dy <= RAD2c);
}

// ---------------------------------------------------------------------------
// CDNA5 async global->LDS copies (ASYNCcnt path)
// ---------------------------------------------------------------------------
__device__ __forceinline__ void async_ld_b128(unsigned lds_addr, const float* gaddr) {
    asm volatile("global_load_async_to_lds_b128 %0, %1, off"
                 :: "v"(lds_addr), "v"(gaddr)
                 : "memory");
}
__device__ __forceinline__ void wait_async0() {
#if __has_builtin(__builtin_amdgcn_s_wait_asynccnt)
    __builtin_amdgcn_s_wait_asynccnt(0);
#else
    asm volatile("s_wait_asynccnt 0" ::: "memory");
#endif
}

// ---------------------------------------------------------------------------
// Fused double Jacobi sweep (temporal blocking, 20 launches for 40 sweeps).
//
// tile0: halo-2 source tile, 12 rows x 40 cols, global cols [x0-4, x0+35]
//        (every 16B chunk aligned in global and LDS), staged with 120
//        global_load_async_to_lds_b128 ops.
// tile1: intermediate sweep p1 over the halo-1 ring [y0-1,y0+8]x[x0-1,x0+32];
//        out-of-domain ring slots hold bc_p(p1) directly (zero-gradient ->
//        value computed at the clamped cell; outlet -> 0).
// Step 2 then updates the JTY x JTX interior from tile1 only.
// ---------------------------------------------------------------------------
constexpr int JTX   = 32;
constexpr int JTY   = 8;
constexpr int JLW   = 40;                 // floats per LDS row
constexpr int JR0   = JTY + 4;            // 12 rows in tile0 (halo 2)
constexpr int JCH   = JLW / 4;            // 10 b128 chunks per row

KERN void jacobi2(const float* __restrict__ psrc, const float* __restrict__ b,
                  float* __restrict__ pdst, float* __restrict__ resOut) {
    __shared__ float tile0[JR0 * JLW];    // 480 floats
    __shared__ float tile1[JR0 * JLW];    // rows 1..10 used
    const int t  = threadIdx.x;
    const int x0 = blockIdx.x * JTX;
    const int y0 = blockIdx.y * JTY;

    // ---- stage halo-2 source tile (rows gy = clamp(y0-2+r)) --------------
    if (t < JR0 * JCH) {                  // 120 lanes issue one b128 each
        int row = t / JCH;
        int c   = t - row * JCH;
        int gy  = min(max(y0 - 2 + row, 0), NYc - 1);
        int gxs = min(max(x0 - 4 + 4 * c, 0), NXc - 4);
        unsigned lds = (unsigned)(uintptr_t)&tile0[row * JLW + 4 * c];
        async_ld_b128(lds, psrc + (size_t)gy * NXc + gxs);
    }
    wait_async0();
    __syncthreads();

    // inlet zero-gradient halo (gx=-1 lives at lx=3)
    if (x0 == 0 && t < JR0)
        tile0[t * JLW + 3] = tile0[t * JLW + 4];
    // outlet block: p=0 for gx >= NX
    if (x0 + JLW - 4 > NXc) {
        for (int li = t; li < JR0 * JLW; li += 256) {
            int lx = li % JLW;
            if (x0 - 4 + lx > NXc - 1) tile0[li] = 0.0f;
        }
    }
    __syncthreads();

    // ---- step 1: p1 over the halo-1 ring (340 slots) ---------------------
    // slot (r=1..10, lx=3..36) labels (gy,gx) = (y0-2+r, x0-4+lx).
    for (int idx = t; idx < 10 * 34; idx += 256) {
        int rr = idx / 34;                // 0..9  -> r  = rr+1
        int cc = idx - rr * 34;           // 0..33 -> lx = cc+3
        int r  = rr + 1;
        int lx = cc + 3;
        int gy = y0 - 2 + r;
        int gx = x0 - 4 + lx;
        float p1;
        if (gx > NXc - 1) {
            p1 = 0.0f;                    // outlet Dirichlet on bc_p(p1)
        } else {
            // zero-gradient edges: evaluate at the clamped cell
            int gyc = min(max(gy, 0), NYc - 1);
            int gxc = max(gx, 0);
            const float* r0 = &tile0[(gyc - y0 + 2 - 1) * JLW + (gxc - x0 + 4 - 1)];
            float s = r0[0]           + r0[1]           + r0[2]
                    + r0[JLW]         + r0[JLW + 1]     + r0[JLW + 2]
                    + r0[2 * JLW]     + r0[2 * JLW + 1] + r0[2 * JLW + 2];
            float c0 = r0[JLW + 1];
            float resid = 3.0f * c0 - s * (1.0f / 3.0f) - b[(size_t)gyc * NXc + gxc];
            p1 = c0 - resid * INV_DIAG;
        }
        tile1[r * JLW + lx] = p1;
    }
    __syncthreads();

    // ---- step 2: final sweep on the JTY x JTX interior -------------------
    const int tx = t & (JTX - 1);
    const int ty = t >> 5;
    const int x = x0 + tx;
    const int y = y0 + ty;
    if (x >= NXc) return;
    const float* r1 = &tile1[(ty + 1) * JLW + tx + 3];
    float s = r1[0]           + r1[1]           + r1[2]
            + r1[JLW]         + r1[JLW + 1]     + r1[JLW + 2]
            + r1[2 * JLW]     + r1[2 * JLW + 1] + r1[2 * JLW + 2];
    float c = r1[JLW + 1];
    float resid = 3.0f * c - s * (1.0f / 3.0f) - b[(size_t)y * NXc + x];
    if (resOut) resOut[(size_t)y * NXc + x] = resid;   // = A*p39 - b on last call
    pdst[(size_t)y * NXc + x] = c - resid * INV_DIAG;
}

// ---------------------------------------------------------------------------
// Helpers for the one-shot kernels
// ---------------------------------------------------------------------------

// Gather with velocity/temperature BCs: left column Dirichlet, edge elsewhere
__device__ __forceinline__ float ld_bc(const float* __restrict__ f, int y, int x,
                                       float leftVal) {
    y = min(max(y, 0), NYc - 1);
    if (x < 0) return leftVal;
    x = min(x, NXc - 1);
    return f[y * NXc + x];
}

// 3x3 stencils (cross-correlation orientation)
__device__ __forceinline__ float st_cx(const float n[3][3]) {
    return ((n[0][0] - n[0][2]) + 4.0f * (n[1][0] - n[1][2]) + (n[2][0] - n[2][2]))
           * (1.0f / 12.0f);
}
__device__ __forceinline__ float st_cy(const float n[3][3]) {
    return ((n[0][0] + 4.0f * n[0][1] + n[0][2])
          - (n[2][0] + 4.0f * n[2][1] + n[2][2])) * (1.0f / 12.0f);
}
__device__ __forceinline__ float st_cd(const float n[3][3]) {
    float s = n[0][0] + n[0][1] + n[0][2]
            + n[1][0] + n[1][1] + n[1][2]
            + n[2][0] + n[2][1] + n[2][2];
    return (s - 9.0f * n[1][1]) * (1.0f / 3.0f);
}

// ---------------------------------------------------------------------------
// One-shot kernels
// ---------------------------------------------------------------------------

// Full padded-array build from an unpadded source (bc_u/v/T: mode 0, bc_p: 1).
KERN void pad_full(const float* __restrict__ src, float* __restrict__ dst,
                   int mode, float leftVal) {
    int i = blockIdx.x * blockDim.x + threadIdx.x;
    if (i >= NPc) return;
    int Y = i / PWc;
    int X = i - Y * PWc;
    float val;
    if (mode == 0 && X == 0)            val = leftVal;
    else if (mode == 1 && X == PWc - 1) val = 0.0f;
    else {
        int yc = min(max(Y - 1, 0), NYc - 1);
        int xc = min(max(X - 1, 0), NXc - 1);
        val = src[yc * NXc + xc];
    }
    dst[i] = val;
}

// Fill only the halo ring of a padded array whose interior is already written.
KERN void edge_fix(float* __restrict__ a, int mode, float leftVal) {
    int i = blockIdx.x * blockDim.x + threadIdx.x;
    const int nb = 2 * PWc + 2 * (PHc - 2);
    if (i >= nb) return;
    int X, Y;
    if (i < PWc)            { Y = 0;       X = i; }
    else if (i < 2 * PWc)   { Y = PHc - 1; X = i - PWc; }
    else {
        int j = i - 2 * PWc;
        if (j < PHc - 2)    { Y = j + 1;             X = 0; }
        else                { Y = j - (PHc - 2) + 1; X = PWc - 1; }
    }
    float val;
    if (mode == 0 && X == 0)            val = leftVal;
    else if (mode == 1 && X == PWc - 1) val = 0.0f;
    else {
        int Yc = min(max(Y, 1), PHc - 2);
        int Xc = min(max(X, 1), PWc - 2);
        val = a[Yc * PWc + Xc];          // reads interior only -> race-free
    }
    a[Y * PWc + X] = val;
}

// Predictor half-step -> interiors of padded outputs b_uu/b_vv/b_TT.
KERN void predictor(const float* __restrict__ u, const float* __restrict__ v,
                    const float* __restrict__ T,
                    float* __restrict__ buu, float* __restrict__ bvv,
                    float* __restrict__ bTT) {
    int x = blockIdx.x * blockDim.x + threadIdx.x;
    int y = blockIdx.y;
    if (x >= NXc) return;
    float un[3][3], vn[3][3], Tn[3][3];
#pragma unroll
    for (int dy = -1; dy <= 1; ++dy)
#pragma unroll
        for (int dx = -1; dx <= 1; ++dx) {
            un[dy + 1][dx + 1] = ld_bc(u, y + dy, x + dx, 1.5f);
            vn[dy + 1][dx + 1] = ld_bc(v, y + dy, x + dx, 0.0f);
            Tn[dy + 1][dx + 1] = ld_bc(T, y + dy, x + dx, 0.0f);
        }
    float uc = un[1][1], vc = vn[1][1], Tc = Tn[1][1];
    bool in = inside_body(x, y);
    float invH = in ? INV_HALF_IN : 1.0f;
    float up = (uc + 0.5f * DTc * (uc * st_cx(un) + vc * st_cy(un) + NUc * st_cd(un))) * invH;
    float vp = (vc + 0.5f * DTc * (uc * st_cx(vn) + vc * st_cy(vn) + NUc * st_cd(vn))) * invH;
    float Tp =  Tc + 0.5f * DTc * (uc * st_cx(Tn) + vc * st_cy(Tn) + KTc * st_cd(Tn));
    Tp = in ? 1.0f : Tp;
    int pi = (y + 1) * PWc + (x + 1);
    buu[pi] = up; bvv[pi] = vp; bTT[pi] = Tp;
}

// Corrector full step: plain 3x3 gathers from BC-complete padded arrays.
KERN void corrector(const float* __restrict__ u, const float* __restrict__ v,
                    const float* __restrict__ T,
                    const float* __restrict__ buu, const float* __restrict__ bvv,
                    const float* __restrict__ bTT,
                    float* __restrict__ uus, float* __restrict__ vvs,
                    float* __restrict__ Tnew) {
    int x = blockIdx.x * blockDim.x + threadIdx.x;
    int y = blockIdx.y;
    if (x >= NXc) return;
    float un[3][3], vn[3][3], Tn[3][3];
#pragma unroll
    for (int dy = 0; dy < 3; ++dy)
#pragma unroll
        for (int dx = 0; dx < 3; ++dx) {
            int idx = (y + dy) * PWc + (x + dx);
            un[dy][dx] = buu[idx];
            vn[dy][dx] = bvv[idx];
            Tn[dy][dx] = bTT[idx];
        }
    float upc = un[1][1], vpc = vn[1][1];
    float uc = u[y * NXc + x], vc = v[y * NXc + x], Tc = T[y * NXc + x];
    bool in = inside_body(x, y);
    float invF = in ? INV_FULL_IN : 1.0f;
    float us = (uc + DTc * (upc * st_cx(un) + vpc * st_cy(un) + NUc * st_cd(un))) * invF;
    float vs = (vc + DTc * (upc * st_cx(vn) + vpc * st_cy(vn) + NUc * st_cd(vn))) * invF;
    float T2 =  Tc + DTc * (upc * st_cx(Tn) + vpc * st_cy(Tn) + KTc * st_cd(Tn));
    T2 = in ? 1.0f : T2;
    int pi = (y + 1) * PWc + (x + 1);
    uus[pi] = us; vvs[pi] = vs;
    Tnew[y * NXc + x] = T2;
}

// Pressure Poisson RHS: b = (cx(bc_u(u_star)) + cy(bc_v(v_star))) / DT
KERN void rhs_kernel(const float* __restrict__ uus, const float* __restrict__ vvs,
                     float* __restrict__ b) {
    int x = blockIdx.x * blockDim.x + threadIdx.x;
    int y = blockIdx.y;
    if (x >= NXc) return;
    float un[3][3], vn[3][3];
#pragma unroll
    for (int dy = 0; dy < 3; ++dy)
#pragma unroll
        for (int dx = 0; dx < 3; ++dx) {
            int idx = (y + dy) * PWc + (x + dx);
            un[dy][dx] = uus[idx];
            vn[dy][dx] = vvs[idx];
        }
    b[y * NXc + x] = (st_cx(un) + st_cy(vn)) * (1.0f / DTc);
}

// Projection: plain gathers from BC-complete pp.
KERN void projection(const float* __restrict__ pp, const float* __restrict__ uus,
                     const float* __restrict__ vvs,
                     float* __restrict__ unew, float* __restrict__ vnew,
                     float* __restrict__ pout) {
    int x = blockIdx.x * blockDim.x + threadIdx.x;
    int y = blockIdx.y;
    if (x >= NXc) return;
    float pn[3][3];
#pragma unroll
    for (int dy = 0; dy < 3; ++dy)
#pragma unroll
        for (int dx = 0; dx < 3; ++dx)
            pn[dy][dx] = pp[(y + dy) * PWc + (x + dx)];
    int pi = (y + 1) * PWc + (x + 1);
    float us = uus[pi], vs = vvs[pi];
    bool in = inside_body(x, y);
    float invF = in ? INV_FULL_IN : 1.0f;
    unew[y * NXc + x] = (us + st_cx(pn) * DTc) * invF;
    vnew[y * NXc + x] = (vs + st_cy(pn) * DTc) * invF;
    pout[y * NXc + x] = pn[1][1];
}

// ---------------------------------------------------------------------------
// Host launch
// ---------------------------------------------------------------------------
extern "C" void kernel_launch(void* const* d_in, const int* in_sizes, int n_in,
                              void* d_out, int out_size, void* d_ws, size_t ws_size,
                              hipStream_t stream) {
    (void)in_sizes; (void)n_in; (void)out_size; (void)ws_size;

    const float* u = (const float*)d_in[0];
    const float* v = (const float*)d_in[1];
    const float* p = (const float*)d_in[2];
    const float* T = (const float*)d_in[3];
    // d_in[4..10] are overwritten scratch in the reference; d_in[11..14] are
    // the fixed stencil weights (hardcoded as constants above).

    float* out    = (float*)d_out;
    float* o_unew = out;
    float* o_vnew = out + (size_t)Nc;
    float* o_p    = out + 2 * (size_t)Nc;
    float* o_Tnew = out + 3 * (size_t)Nc;
    float* o_uus  = out + 4 * (size_t)Nc;
    float* o_vvs  = o_uus + (size_t)NPc;
    float* o_pp   = o_uus + 2 * (size_t)NPc;
    float* o_TT   = o_uus + 3 * (size_t)NPc;
    float* o_buu  = o_uus + 4 * (size_t)NPc;
    float* o_bvv  = o_uus + 5 * (size_t)NPc;
    float* o_bTT  = o_uus + 6 * (size_t)NPc;
    float* o_res  = o_uus + 7 * (size_t)NPc;

    float* ws   = (float*)d_ws;
    float* w_b  = ws;                      // Poisson RHS
    float* w_pA = ws + (size_t)Nc;         // Jacobi ping
    float* w_pB = ws + 2 * (size_t)Nc;     // Jacobi pong

    const dim3 blk(256);
    const dim3 grd2((NXc + 255) / 256, NYc);
    const int  nb = 2 * PWc + 2 * (PHc - 2);
    const dim3 grdB((nb + 255) / 256);
    const dim3 grdP((NPc + 255) / 256);
    const dim3 grdJ((NXc + JTX - 1) / JTX, NYc / JTY);

    // TT output = bc_T(original T)
    pad_full<<<grdP, blk, 0, stream>>>(T, o_TT, 0, 0.0f);

    // Predictor -> b_uu/b_vv/b_TT interiors, then BC halos
    predictor<<<grd2, blk, 0, stream>>>(u, v, T, o_buu, o_bvv, o_bTT);
    edge_fix<<<grdB, blk, 0, stream>>>(o_buu, 0, 1.5f);
    edge_fix<<<grdB, blk, 0, stream>>>(o_bvv, 0, 0.0f);
    edge_fix<<<grdB, blk, 0, stream>>>(o_bTT, 0, 0.0f);

    // Corrector -> uu_star/vv_star interiors + T_new, then BC halos
    corrector<<<grd2, blk, 0, stream>>>(u, v, T, o_buu, o_bvv, o_bTT,
                                        o_uus, o_vvs, o_Tnew);
    edge_fix<<<grdB, blk, 0, stream>>>(o_uus, 0, 1.5f);
    edge_fix<<<grdB, blk, 0, stream>>>(o_vvs, 0, 0.0f);

    // Poisson RHS
    rhs_kernel<<<grd2, blk, 0, stream>>>(o_uus, o_vvs, w_b);

    // 40 Jacobi sweeps as 20 fused double sweeps (ping-pong);
    // the last fused kernel emits the 40th-sweep residual (A*p39 - b).
    const float* src = p;
    float* dst = w_pA;
    for (int k = 1; k <= 20; ++k) {
        float* r = (k == 20) ? o_res : nullptr;
        jacobi2<<<grdJ, blk, 0, stream>>>(src, w_b, dst, r);
        src = dst;
        dst = (dst == w_pA) ? w_pB : w_pA;
    }
    const float* pfinal = src;

    // pp output = bc_p(p_final), then projection
    pad_full<<<grdP, blk, 0, stream>>>(pfinal, o_pp, 1, 0.0f);
    projection<<<grd2, blk, 0, stream>>>(o_pp, o_uus, o_vvs,
                                         o_unew, o_vnew, o_p);
}